// DMLNegHead_36910948942208
// MI455X (gfx1250) — compile-verified
//
#include <hip/hip_runtime.h>
#include <math.h>

typedef float v2f __attribute__((ext_vector_type(2)));
typedef float v8f __attribute__((ext_vector_type(8)));

#define NB    16
#define NC    256
#define NE    256
#define NHW   4096
#define NP    65536          // B*H*W
#define NR    365
#define NRP   368            // padded to multiple of 16
#define NRT   23             // NRP/16
#define S_OUT 23920640       // 16*365*4096
#define PRS   96             // pair-row stride in dwords (== 32 mod 64 -> bank split)

static __device__ __forceinline__ v8f wmma_f32(v2f a, v2f b, v8f c) {
    return __builtin_amdgcn_wmma_f32_16x16x4_f32(false, a, false, b, (short)0, c, false, false);
}

// ---------------------------------------------------------------------------
// Kernel A: build padded rep matrices + squared norms.
// ---------------------------------------------------------------------------
__device__ float block_reduce_sum256(float v, float* red) {
    int t = threadIdx.x;
    red[t] = v;
    __syncthreads();
    for (int s = 128; s > 0; s >>= 1) {
        if (t < s) red[t] += red[t + s];
        __syncthreads();
    }
    float r = red[0];
    __syncthreads();
    return r;
}

__global__ void reps_prep_kernel(const float* __restrict__ reps,
                                 const float* __restrict__ neg_w,
                                 const float* __restrict__ neg_b,
                                 float* __restrict__ Rpos,
                                 float* __restrict__ Rneg0,
                                 float* __restrict__ Rneg1,
                                 float* __restrict__ rn2p,
                                 float* __restrict__ rn2n0,
                                 float* __restrict__ rn2n1) {
    int r = blockIdx.x;
    int t = threadIdx.x;
    __shared__ float absr[NE];
    __shared__ float red[256];

    if (r >= NR) {  // zero padding rows so WMMA tiles are regular
        Rpos [r * NE + t] = 0.f;
        Rneg0[r * NE + t] = 0.f;
        Rneg1[r * NE + t] = 0.f;
        if (t == 0) { rn2p[r] = 0.f; rn2n0[r] = 0.f; rn2n1[r] = 0.f; }
        return;
    }

    float rv = reps[r * NE + t];       // representations[r,0,t]
    float av = fabsf(rv);
    absr[t] = av;
    __syncthreads();

    // neg_offset[r, m, t] = sum_e |reps[r,e]| * neg_w[m*E + t, e] + neg_b[m*E + t]
    float o0 = neg_b[t];
    float o1 = neg_b[NE + t];
    const float* w0 = neg_w + (size_t)t * NE;
    const float* w1 = neg_w + (size_t)(NE + t) * NE;
    for (int e = 0; e < NE; ++e) {
        float a = absr[e];
        o0 = fmaf(a, w0[e], o0);
        o1 = fmaf(a, w1[e], o1);
    }

    float sg = (rv > 0.f) ? 1.f : ((rv < 0.f) ? -1.f : 0.f);
    float v0 = (o0 + av) * sg;
    float v1 = (o1 + av) * sg;

    float sp = block_reduce_sum256(rv * rv, red);
    float s0 = block_reduce_sum256(v0 * v0, red);
    float s1 = block_reduce_sum256(v1 * v1, red);

    float i0 = 1.f / fmaxf(sqrtf(s0), 1e-12f);
    float i1 = 1.f / fmaxf(sqrtf(s1), 1e-12f);

    Rpos [r * NE + t] = rv;
    Rneg0[r * NE + t] = v0 * i0;
    Rneg1[r * NE + t] = v1 * i1;
    if (t == 0) {
        rn2p [r] = sp;               // reps already normalized at setup
        rn2n0[r] = s0 * i0 * i0;     // = ||normalized||^2 (≈1, kept exact)
        rn2n1[r] = s1 * i1 * i1;
    }
}

// ---------------------------------------------------------------------------
// Kernel B: emb GEMM (A = conv_w [E x C], B = x tile [C x 32px]) via f32 WMMA.
// 64 threads (2 waves) per block; waves split the 16 e-tiles. The x tile is
// staged in pair-interleaved LDS so every B operand is one ds_load_b64.
// Writes raw embT [E][P] + per-pixel inv_norm and en2.
// ---------------------------------------------------------------------------
__global__ __launch_bounds__(64) void emb_gemm_kernel(
    const float* __restrict__ x,
    const float* __restrict__ conv_w,
    const float* __restrict__ conv_b,
    float* __restrict__ embT,
    float* __restrict__ inv_norm,
    float* __restrict__ en2_out) {
    __shared__ __align__(16) float Bsh[(NC / 2) * PRS];  // pair-interleaved, 48 KB
    __shared__ float red[64];

    const int tid  = threadIdx.x;
    const int wave = tid >> 5;
    const int l    = tid & 31;
    const int n    = l & 15;
    const int half = l >> 4;
    const int p0   = blockIdx.x * 32;
    const int b    = p0 >> 12;
    const int hw0  = p0 & (NHW - 1);

    // stride-64 staging: (tid + 64t) & 31 == l, so each lane owns column l.
    // lds[(c>>1)*PRS + l*2 + (c&1)]  <- x[c][pixel l]
    const float* xb = x + (size_t)b * NC * NHW + hw0;
    for (int idx = tid; idx < NC * 32; idx += 64) {
        int c = idx >> 5;
        Bsh[(c >> 1) * PRS + l * 2 + (c & 1)] = xb[(size_t)c * NHW + l];
    }
    __syncthreads();

    float ssq0 = 0.f, ssq1 = 0.f;
    for (int e0 = wave * 16; e0 < NE; e0 += 32) {
        v8f acc0 = {0.f,0.f,0.f,0.f,0.f,0.f,0.f,0.f};
        v8f acc1 = {0.f,0.f,0.f,0.f,0.f,0.f,0.f,0.f};
        const float* Arow = conv_w + (size_t)(e0 + n) * NC;   // lane m = l&15
        for (int k0 = 0; k0 < NC; k0 += 4) {
            int pr = (k0 >> 1) + half;        // pair-row for this half-wave
            const float* bp = &Bsh[pr * PRS + n * 2];
            v2f a;  a.x = Arow[k0 + 2 * half]; a.y = Arow[k0 + 2 * half + 1];
            v2f b0 = *(const v2f*)(bp);       // one ds_load_b64
            v2f b1 = *(const v2f*)(bp + 32);
            acc0 = wmma_f32(a, b0, acc0);
            acc1 = wmma_f32(a, b1, acc1);
        }
#pragma unroll
        for (int j = 0; j < 8; ++j) {
            int e = e0 + j + 8 * half;        // C layout: m = j + 8*(lane/16)
            float bias = conv_b[e];
            float v0 = acc0[j] + bias;
            float v1 = acc1[j] + bias;
            embT[(size_t)e * NP + p0 + n]      = v0;
            embT[(size_t)e * NP + p0 + 16 + n] = v1;
            ssq0 += v0 * v0;
            ssq1 += v1 * v1;
        }
    }
    ssq0 += __shfl_xor(ssq0, 16, 32);          // combine the two e-halves
    ssq1 += __shfl_xor(ssq1, 16, 32);
    if (l < 16) {
        red[wave * 32 + n]      = ssq0;
        red[wave * 32 + 16 + n] = ssq1;
    }
    __syncthreads();
    if (tid < 32) {
        float ssq = red[tid] + red[32 + tid];  // both waves' e-tile partials
        float inv = 1.f / fmaxf(sqrtf(ssq), 1e-12f);
        inv_norm[p0 + tid] = inv;
        en2_out [p0 + tid] = ssq * inv * inv;
    }
}

// ---------------------------------------------------------------------------
// Kernel C: distance GEMM against all 365 pos + 2x365 neg reps + full
// post-processing. 64-thread blocks (2 waves) own a 32-pixel tile staged once
// in pair-interleaved LDS; the 23 rep tiles are split between the waves so the
// rep matrices are read once per block and each A load feeds two WMMAs.
// Per-pixel sum over r is reduced deterministically (shuffle + LDS).
// ---------------------------------------------------------------------------
__global__ __launch_bounds__(64) void dist_kernel(
    const float* __restrict__ embT,
    const float* __restrict__ inv_norm,
    const float* __restrict__ en2_in,
    const float* __restrict__ Rpos,
    const float* __restrict__ Rneg0,
    const float* __restrict__ Rneg1,
    const float* __restrict__ rn2p,
    const float* __restrict__ rn2n0,
    const float* __restrict__ rn2n1,
    float* __restrict__ out_cls,      // unnormalized probs (kernel D divides)
    float* __restrict__ out_clsneg,
    float* __restrict__ out_d,
    float* __restrict__ out_dn,
    float* __restrict__ out_pori,
    float* __restrict__ rsum) {
    __shared__ __align__(16) float Bsh[(NE / 2) * PRS];  // pair-interleaved, 48 KB
    __shared__ float red[64];        // cross-wave per-pixel partial sums

    const int tid  = threadIdx.x;
    const int wave = tid >> 5;
    const int l    = tid & 31;
    const int n    = l & 15;
    const int half = l >> 4;
    const int p0   = blockIdx.x * 32;
    const int b    = p0 >> 12;
    const int hw0  = p0 & (NHW - 1);

    // stride-64 staging: each lane owns column l; scale by 1/||emb|| here.
    const float invn = inv_norm[p0 + l];
    for (int idx = tid; idx < NE * 32; idx += 64) {
        int k = idx >> 5;
        Bsh[(k >> 1) * PRS + l * 2 + (k & 1)] =
            embT[(size_t)k * NP + p0 + l] * invn;
    }
    __syncthreads();

    const float en2q[2] = { en2_in[p0 + n], en2_in[p0 + 16 + n] };
    const float c2 = 2.0f;   // 1/(2*sigma^2), sigma = 0.5
    float rs[2] = {0.f, 0.f};

    for (int rt = wave; rt < NRT; rt += 2) {
        int r0 = rt * 16;
        v8f accP[2], accN0[2], accN1[2];
#pragma unroll
        for (int q = 0; q < 2; ++q) {
            accP[q]  = (v8f){0.f,0.f,0.f,0.f,0.f,0.f,0.f,0.f};
            accN0[q] = (v8f){0.f,0.f,0.f,0.f,0.f,0.f,0.f,0.f};
            accN1[q] = (v8f){0.f,0.f,0.f,0.f,0.f,0.f,0.f,0.f};
        }
        const float* Ap = Rpos  + (size_t)(r0 + n) * NE;
        const float* A0 = Rneg0 + (size_t)(r0 + n) * NE;
        const float* A1 = Rneg1 + (size_t)(r0 + n) * NE;
        for (int k0 = 0; k0 < NE; k0 += 4) {
            int ka = k0 + 2 * half;
            int pr = (k0 >> 1) + half;
            const float* bp = &Bsh[pr * PRS + n * 2];
            v2f b0 = *(const v2f*)(bp);        // one ds_load_b64 each
            v2f b1 = *(const v2f*)(bp + 32);
            v2f ap; ap.x = Ap[ka];  ap.y = Ap[ka + 1];
            v2f a0; a0.x = A0[ka];  a0.y = A0[ka + 1];
            v2f a1; a1.x = A1[ka];  a1.y = A1[ka + 1];
            accP[0]  = wmma_f32(ap, b0, accP[0]);
            accP[1]  = wmma_f32(ap, b1, accP[1]);
            accN0[0] = wmma_f32(a0, b0, accN0[0]);
            accN0[1] = wmma_f32(a0, b1, accN0[1]);
            accN1[0] = wmma_f32(a1, b0, accN1[0]);
            accN1[1] = wmma_f32(a1, b1, accN1[1]);
        }
#pragma unroll
        for (int q = 0; q < 2; ++q) {
            float en2 = en2q[q];
#pragma unroll
            for (int j = 0; j < 8; ++j) {
                int r = r0 + j + 8 * half;
                float d2  = fmaxf(en2 + rn2p [r] - 2.f * accP[q][j],  0.f);
                float d   = sqrtf(d2);
                float dq0 = fmaxf(en2 + rn2n0[r] - 2.f * accN0[q][j], 0.f);
                float dn0 = sqrtf(dq0);
                float dq1 = fmaxf(en2 + rn2n1[r] - 2.f * accN1[q][j], 0.f);
                float dn1 = sqrtf(dq1);
                float dnm   = fminf(dn0, dn1);
                float csneg = expf(-dnm * dnm * c2);   // max_m exp(-dn^2*c)
                float pori  = expf(-d * d * c2);
                float t     = d + 0.3f * (2.f - dnm);
                float probs = expf(-t * t * c2);
                if (r < NR) {
                    size_t base  = ((size_t)b * NR + r) * NHW + hw0 + q * 16 + n;
                    size_t basen = (((size_t)b * NR + r) * 2) * NHW + hw0 + q * 16 + n;
                    out_cls   [base]        = probs;
                    out_clsneg[base]        = csneg;
                    out_d     [base]        = d;
                    out_dn    [basen]       = dn0;
                    out_dn    [basen + NHW] = dn1;
                    out_pori  [base]        = pori;
                    rs[q] += probs;
                }
            }
        }
    }
    rs[0] += __shfl_xor(rs[0], 16, 32);     // two r-halves share a pixel
    rs[1] += __shfl_xor(rs[1], 16, 32);
    if (l < 16) {
        red[wave * 32 + n]      = rs[0];
        red[wave * 32 + 16 + n] = rs[1];
    }
    __syncthreads();
    if (tid < 32) rsum[p0 + tid] = red[tid] + red[32 + tid];
}

// ---------------------------------------------------------------------------
// Kernel D: cls_score normalization (divide by per-pixel sum over r).
// ---------------------------------------------------------------------------
__global__ void cls_norm_kernel(float* __restrict__ out_cls,
                                const float* __restrict__ rsum) {
    int idx = blockIdx.x * 256 + threadIdx.x;
    if (idx >= S_OUT) return;
    int hw = idx & (NHW - 1);
    int b  = idx / (NR * NHW);
    out_cls[idx] = out_cls[idx] / rsum[b * NHW + hw];
}

// ---------------------------------------------------------------------------
extern "C" void kernel_launch(void* const* d_in, const int* in_sizes, int n_in,
                              void* d_out, int out_size, void* d_ws, size_t ws_size,
                              hipStream_t stream) {
    (void)in_sizes; (void)n_in; (void)out_size; (void)ws_size;

    const float* x      = (const float*)d_in[0];
    const float* conv_w = (const float*)d_in[1];
    const float* conv_b = (const float*)d_in[2];
    const float* reps   = (const float*)d_in[3];
    const float* neg_w  = (const float*)d_in[4];
    const float* neg_b  = (const float*)d_in[5];

    // workspace layout (floats): ~69 MB total
    float* ws       = (float*)d_ws;
    float* Rpos     = ws;                         // 368*256
    float* Rneg0    = Rpos  + NRP * NE;
    float* Rneg1    = Rneg0 + NRP * NE;
    float* rn2p     = Rneg1 + NRP * NE;           // 368 (padded slots of 384)
    float* rn2n0    = rn2p  + 384;
    float* rn2n1    = rn2n0 + 384;
    float* inv_norm = rn2n1 + 384;                // 65536
    float* en2      = inv_norm + NP;
    float* rsum     = en2 + NP;
    float* embT     = rsum + NP;                  // 256*65536

    // outputs, concatenated in reference return order
    float* out_cls    = (float*)d_out;            // S
    float* out_clsneg = out_cls    + (size_t)S_OUT;
    float* out_d      = out_clsneg + (size_t)S_OUT;
    float* out_dn     = out_d      + (size_t)S_OUT;   // 2S
    float* out_pori   = out_dn     + (size_t)2 * S_OUT;

    reps_prep_kernel<<<NRP, 256, 0, stream>>>(reps, neg_w, neg_b,
                                              Rpos, Rneg0, Rneg1,
                                              rn2p, rn2n0, rn2n1);

    emb_gemm_kernel<<<NP / 32, 64, 0, stream>>>(x, conv_w, conv_b,
                                                embT, inv_norm, en2);

    dist_kernel<<<NP / 32, 64, 0, stream>>>(embT, inv_norm, en2,
                                            Rpos, Rneg0, Rneg1,
                                            rn2p, rn2n0, rn2n1,
                                            out_cls, out_clsneg, out_d,
                                            out_dn, out_pori, rsum);

    cls_norm_kernel<<<(S_OUT + 255) / 256, 256, 0, stream>>>(out_cls, rsum);
}